// DyanEncoder_81690277970358
// MI455X (gfx1250) — compile-verified
//
#include <hip/hip_runtime.h>
#include <math.h>

typedef __attribute__((ext_vector_type(2))) float v2f;
typedef __attribute__((ext_vector_type(8))) float v8f;

#define T_FRAMES 36
#define NPOLE    80
#define P        161          // 2*80+1
#define PP       176          // padded to 11*16
#define NB       64
#define DD       384
#define NC       (NB*DD)      // 24576 columns
#define LAMBDA   0.1f
#define MAX_ITER 100
#define THRESH2  1.4745600e-05f   // (1e-5 * 384)^2

// workspace layout (float offsets)
#define OFF_DPAD 0                         // [48][176] padded dictionary (row t, col p)
#define OFF_DTD  8448                      // [161][161]
#define OFF_APAD 34432                     // [176][176] A = I - Linv*DtD (zero padded)
#define OFF_B    65536                     // [176][24576]
#define OFF_X    (OFF_B + PP*NC)
#define OFF_Y    (OFF_X + PP*NC)
#define OFF_MISC (OFF_Y + PP*NC)
#define MISC_LINV 0
#define MISC_TT   16                       // tt[k], k<100
#define MISC_PART 160                      // per-block partial sums (192)
#define MISC_FLAG 384                      // int done flags [101]

// output layout: (C, dic, reconst) flat
#define OUT_DIC  (NB*P*DD)                 // 3,956,736
#define OUT_REC  (OUT_DIC + T_FRAMES*P)    // +5,796

__device__ __forceinline__ v8f wmma_f32(v2f a, v2f b, v8f c) {
  return __builtin_amdgcn_wmma_f32_16x16x4_f32(false, a, false, b, (short)0, c,
                                               false, false);
}

// ---------------------------------------------------------------- dictionary
__global__ void k_build_dict(const float* __restrict__ rr,
                             const float* __restrict__ th,
                             float* __restrict__ ws,
                             float* __restrict__ out) {
  int idx = blockIdx.x * blockDim.x + threadIdx.x;   // covers 48*176 = 8448
  if (idx >= 48 * PP) return;
  int t = idx / PP, p = idx % PP;
  float v = 0.f;
  if (t < T_FRAMES && p < P) {
    float ft = (float)t;
    if (p == 0) v = 1.f;
    else if (p <= NPOLE) { int j = p - 1;         v = powf(rr[j], ft) * cosf(ft * th[j]); }
    else                 { int j = p - 1 - NPOLE; v = powf(rr[j], ft) * sinf(ft * th[j]); }
    out[OUT_DIC + t * P + p] = v;
  }
  ws[OFF_DPAD + t * PP + p] = v;
}

// ---------------------------------------------------------------- DtD
__global__ void k_dtd(float* __restrict__ ws) {
  int idx = blockIdx.x * blockDim.x + threadIdx.x;
  if (idx >= P * P) return;
  int p = idx / P, q = idx % P;
  const float* D = ws + OFF_DPAD;
  float s = 0.f;
  for (int t = 0; t < T_FRAMES; ++t) s += D[t * PP + p] * D[t * PP + q];
  ws[OFF_DTD + idx] = s;
}

// ------------------------------------------- power iteration + tt table + flags
__global__ void k_power(float* __restrict__ ws) {
  __shared__ float pv[P];
  __shared__ float pw[P];
  __shared__ float pn;
  int tid = threadIdx.x;
  const float* M = ws + OFF_DTD;
  if (tid < P) pv[tid] = 1.f;
  __syncthreads();
  for (int it = 0; it < 256; ++it) {
    if (tid < P) {
      float w = 0.f;
      for (int q = 0; q < P; ++q) w += M[tid * P + q] * pv[q];
      pw[tid] = w;
    }
    __syncthreads();
    if (tid == 0) {
      float s = 0.f;
      for (int q = 0; q < P; ++q) s += pw[q] * pw[q];
      pn = sqrtf(s);
    }
    __syncthreads();
    if (tid < P) pv[tid] = pw[tid] / pn;
    __syncthreads();
  }
  if (tid == 0) {
    ws[OFF_MISC + MISC_LINV] = 1.f / pn;     // 1 / spectral norm (SPD -> lambda_max)
    float t = 1.f;
    for (int k = 0; k < MAX_ITER; ++k) {     // deterministic momentum sequence
      float tn = 0.5f * (1.f + sqrtf(1.f + 4.f * t * t));
      ws[OFF_MISC + MISC_TT + k] = (t - 1.f) / tn;
      t = tn;
    }
  }
  int* flags = (int*)(ws + OFF_MISC + MISC_FLAG);
  for (int i = tid; i <= MAX_ITER; i += blockDim.x) flags[i] = 0;
}

// ---------------------------------------------------------------- A = I - Linv*DtD
__global__ void k_build_A(float* __restrict__ ws) {
  int idx = blockIdx.x * blockDim.x + threadIdx.x;   // 176*176 = 30976
  if (idx >= PP * PP) return;
  int p = idx / PP, q = idx % PP;
  float v = 0.f;
  if (p < P && q < P) {
    float linv = ws[OFF_MISC + MISC_LINV];
    v = ((p == q) ? 1.f : 0.f) - linv * ws[OFF_DTD + p * P + q];
  }
  ws[OFF_APAD + idx] = v;
}

// ------------------------------------- B = Linv * D^T Y  (WMMA, K=36), init x=y=0
__global__ void k_compute_B(const float* __restrict__ Y, float* __restrict__ ws) {
  int lane = threadIdx.x & 31, wv = threadIdx.x >> 5;
  int ct = blockIdx.x * 8 + wv;              // 0..1535 column tiles of 16
  int col = ct * 16 + (lane & 15);
  int n = ct / 24;
  int dcol = col - n * DD;
  int kbh = (lane >> 4) << 1;                // 0 or 2
  const float* Dp = ws + OFF_DPAD;
  v8f acc[11] = {};
  for (int kt = 0; kt < 9; ++kt) {           // K = 36 = 9*4
    int k0 = kt * 4 + kbh;
    v2f b;
    b.x = Y[(n * T_FRAMES + k0) * DD + dcol];
    b.y = Y[(n * T_FRAMES + k0 + 1) * DD + dcol];
    for (int mt = 0; mt < 11; ++mt) {
      int m = mt * 16 + (lane & 15);
      v2f a;                                 // A-operand = D^T : a[m][k] = D[k][m]
      a.x = Dp[k0 * PP + m];
      a.y = Dp[(k0 + 1) * PP + m];
      acc[mt] = wmma_f32(a, b, acc[mt]);
    }
  }
  float linv = ws[OFF_MISC + MISC_LINV];
  int rbase = (lane >> 4) << 3;
  for (int mt = 0; mt < 11; ++mt)
    for (int r = 0; r < 8; ++r) {
      int row = mt * 16 + rbase + r;
      size_t id = (size_t)row * NC + col;
      ws[OFF_B + id] = linv * acc[mt][r];
      ws[OFF_X + id] = 0.f;
      ws[OFF_Y + id] = 0.f;
    }
}

// --------------------------------------------- one FISTA iteration (WMMA core)
__global__ void k_step(float* __restrict__ ws, int k) {
  extern __shared__ float sm[];
  float* sA  = sm;              // 176*176 staged A matrix (121 KB of the 320KB WGP LDS)
  float* red = sm + PP * PP;    // 8 per-wave partials
  const int* flags = (const int*)(ws + OFF_MISC + MISC_FLAG);
  if (flags[k]) return;         // frozen: exact early-exit (uniform across grid)

  int tid = threadIdx.x;
  for (int i = tid; i < PP * PP; i += blockDim.x) sA[i] = ws[OFF_APAD + i];
  __syncthreads();

  int lane = tid & 31, wv = tid >> 5;
  int ct = blockIdx.x * 8 + wv;
  int col = ct * 16 + (lane & 15);
  int kbh = (lane >> 4) << 1;
  int rbase = (lane >> 4) << 3;

  v8f acc[11];                               // z starts as B (C-operand layout)
  for (int mt = 0; mt < 11; ++mt)
    for (int r = 0; r < 8; ++r)
      acc[mt][r] = ws[OFF_B + (size_t)(mt * 16 + rbase + r) * NC + col];

  const float* Yw = ws + OFF_Y;
  for (int kt = 0; kt < 44; ++kt) {          // K = 176 = 44*4
    int k0 = kt * 4 + kbh;
    v2f b;
    b.x = Yw[(size_t)k0 * NC + col];
    b.y = Yw[(size_t)(k0 + 1) * NC + col];
    for (int mt = 0; mt < 11; ++mt) {
      float2 av = *(const float2*)&sA[(mt * 16 + (lane & 15)) * PP + k0];
      v2f a; a.x = av.x; a.y = av.y;
      acc[mt] = wmma_f32(a, b, acc[mt]);
    }
  }

  float linv = ws[OFF_MISC + MISC_LINV];
  float thr  = LAMBDA * linv;
  float tt   = ws[OFF_MISC + MISC_TT + k];
  float sq = 0.f;
  for (int mt = 0; mt < 11; ++mt)
    for (int r = 0; r < 8; ++r) {
      int row = mt * 16 + rbase + r;
      size_t id = (size_t)row * NC + col;
      float z  = acc[mt][r];
      float xn = copysignf(fmaxf(fabsf(z) - thr, 0.f), z);   // softshrink
      float xo = ws[OFF_X + id];
      float dl = xo - xn;
      sq += dl * dl;
      ws[OFF_X + id] = xn;
      ws[OFF_Y + id] = (1.f + tt) * xn - tt * xo;            // momentum
    }

  // deterministic block reduction of ||x - x_new||^2
  for (int off = 16; off > 0; off >>= 1) sq += __shfl_down(sq, off, 32);
  if (lane == 0) red[wv] = sq;
  __syncthreads();
  if (tid == 0) {
    float s = 0.f;
    for (int w = 0; w < 8; ++w) s += red[w];
    ws[OFF_MISC + MISC_PART + blockIdx.x] = s;
  }
}

// ------------------------- fixed-order global reduction -> done flag for iter k+1
__global__ void k_reduce(float* __restrict__ ws, int k) {
  if (threadIdx.x != 0) return;
  int* flags = (int*)(ws + OFF_MISC + MISC_FLAG);
  int done = flags[k];
  if (!done) {
    const float* part = ws + OFF_MISC + MISC_PART;
    float s = 0.f;
    for (int i = 0; i < 192; ++i) s += part[i];
    if (s < THRESH2) done = 1;               // sqrt(s)/384 < 1e-5
  }
  flags[k + 1] = done;
}

// ---------------------------------------------------------------- C output (transpose)
__global__ void k_copy_C(const float* __restrict__ ws, float* __restrict__ out) {
  size_t idx = (size_t)blockIdx.x * blockDim.x + threadIdx.x;  // 3,956,736 exact
  int n = (int)(idx / (P * DD));
  int rem = (int)(idx % (P * DD));
  int p = rem / DD, d = rem % DD;
  out[idx] = ws[OFF_X + (size_t)p * NC + n * DD + d];
}

// ---------------------------------------------------------------- reconst = D @ C
__global__ void k_reconst(const float* __restrict__ ws, float* __restrict__ out) {
  int lane = threadIdx.x & 31, wv = threadIdx.x >> 5;
  int ct = blockIdx.x * 8 + wv;
  int col = ct * 16 + (lane & 15);
  int n = ct / 24;
  int dcol = col - n * DD;
  int kbh = (lane >> 4) << 1;
  const float* Xw = ws + OFF_X;
  v8f acc[3] = {};                           // M = 48 rows (36 valid)
  for (int kt = 0; kt < 44; ++kt) {
    int k0 = kt * 4 + kbh;
    v2f b;
    b.x = Xw[(size_t)k0 * NC + col];
    b.y = Xw[(size_t)(k0 + 1) * NC + col];
    for (int mt = 0; mt < 3; ++mt) {
      float2 av = *(const float2*)&ws[OFF_DPAD + (mt * 16 + (lane & 15)) * PP + k0];
      v2f a; a.x = av.x; a.y = av.y;
      acc[mt] = wmma_f32(a, b, acc[mt]);
    }
  }
  int rbase = (lane >> 4) << 3;
  for (int mt = 0; mt < 3; ++mt)
    for (int r = 0; r < 8; ++r) {
      int t = mt * 16 + rbase + r;
      if (t < T_FRAMES)
        out[OUT_REC + ((size_t)n * T_FRAMES + t) * DD + dcol] = acc[mt][r];
    }
}

extern "C" void kernel_launch(void* const* d_in, const int* in_sizes, int n_in,
                              void* d_out, int out_size, void* d_ws, size_t ws_size,
                              hipStream_t stream) {
  (void)in_sizes; (void)n_in; (void)out_size; (void)ws_size;
  const float* x  = (const float*)d_in[0];   // [64,36,384]
  const float* rr = (const float*)d_in[1];   // [80]
  const float* th = (const float*)d_in[2];   // [80]
  float* out = (float*)d_out;
  float* ws  = (float*)d_ws;

  k_build_dict<<<33, 256, 0, stream>>>(rr, th, ws, out);
  k_dtd<<<(P * P + 255) / 256, 256, 0, stream>>>(ws);
  k_power<<<1, 256, 0, stream>>>(ws);
  k_build_A<<<PP * PP / 256, 256, 0, stream>>>(ws);
  k_compute_B<<<192, 256, 0, stream>>>(x, ws);

  size_t shbytes = (size_t)(PP * PP + 8) * sizeof(float);
  for (int k = 0; k < MAX_ITER; ++k) {
    k_step<<<192, 256, shbytes, stream>>>(ws, k);
    k_reduce<<<1, 32, 0, stream>>>(ws, k);
  }

  k_copy_C<<<15456, 256, 0, stream>>>(ws, out);
  k_reconst<<<192, 256, 0, stream>>>(ws, out);
}